// _KCnnLayer_16707422781891
// MI455X (gfx1250) — compile-verified
//
#include <hip/hip_runtime.h>

// CDNA5 / gfx1250 — wave32, WMMA f32 16x16x4.
typedef __attribute__((ext_vector_type(2))) float v2f;
typedef __attribute__((ext_vector_type(4))) float v4f;
typedef __attribute__((ext_vector_type(8))) float v8f;

#define BATCH   32
#define SEQ_N   2000
#define KNBR    32
#define FEAT    64
#define OUTF    128
#define KDIM    (KNBR * (FEAT + 1))   // 2080
#define NCHUNK  8                     // neighbors processed 4 at a time
#define KCH     (KDIM / NCHUNK)       // 260 floats per K-chunk (divisible by 4)
#define MT      16                    // M-tile rows (one WMMA tile)

__global__ __launch_bounds__(256) void kcnn_wmma_f32(
    const float* __restrict__ x,      // (B, N, F)
    const int*   __restrict__ nbh,    // (N, K)
    const float* __restrict__ dist,   // (N, K)
    const float* __restrict__ W,      // (OUT, K*(F+1)) row-major
    const float* __restrict__ bias,   // (OUT,)
    float*       __restrict__ out)    // (B, N, OUT)
{
    __shared__ float hsh[MT * KCH];   // 16 * 260 * 4 = 16,640 B

    const int tid   = threadIdx.x;
    const int lane  = tid & 31;
    const int wave  = tid >> 5;          // 0..7 -> OUT tile
    const int n0    = blockIdx.x * MT;   // n-tile base (125 tiles)
    const int bidx  = blockIdx.y;        // batch index
    const int obase = wave * 16;
    const int l15   = lane & 15;
    const int hi    = lane >> 4;         // 0 for lanes 0-15, 1 for 16-31

    // C accumulator: for a given lane, all 8 VGPRs hold the same output
    // column (N = lane), rows M = r + 8*hi. Initialize with bias.
    const float bv = bias[obase + l15];
    v8f acc;
    #pragma unroll
    for (int i = 0; i < 8; ++i) acc[i] = bv;

    // B-matrix source row for this lane (W is L2-resident: 1.06 MB).
    const float* Wrow = W + (size_t)(obase + l15) * KDIM;

    // Staging decomposition: 64 (row, neighbor) segments x 4 threads.
    const int seg = tid >> 2;   // 0..63
    const int sub = tid & 3;    // 0..3 -> which 16-float slab of the 64 feats
    const int m   = seg >> 2;   // 0..15 local row
    const int j   = seg & 3;    // 0..3  neighbor within chunk
    const int n   = n0 + m;

    for (int c = 0; c < NCHUNK; ++c) {
        __syncthreads();  // previous chunk fully consumed
        // ---- stage h chunk into LDS: neighbors kk in [4c, 4c+4) ----
        {
            const int kk  = 4 * c + j;
            const int nbr = nbh[n * KNBR + kk];
            const float* src = x + ((size_t)bidx * SEQ_N + (size_t)nbr) * FEAT
                                 + sub * 16;
            float* dstp = &hsh[m * KCH + j * (FEAT + 1) + sub * 16];
            const v4f* s4 = reinterpret_cast<const v4f*>(src);  // 64B aligned
            #pragma unroll
            for (int q = 0; q < 4; ++q) {
                v4f v = s4[q];
                dstp[4 * q + 0] = v.x;
                dstp[4 * q + 1] = v.y;
                dstp[4 * q + 2] = v.z;
                dstp[4 * q + 3] = v.w;
            }
            if (sub == 0)
                hsh[m * KCH + j * (FEAT + 1) + FEAT] = dist[n * KNBR + kk];
        }
        __syncthreads();

        // ---- compute: 65 K-steps of 4 over this chunk ----
        const float* wr = Wrow + c * KCH;
        for (int s = 0; s < KCH / 4; ++s) {
            const int kc = 4 * s + 2 * hi;
            // A: 16x4 f32 -> lane L<16 holds (M=L, K=kc..kc+1),
            //                lane L+16 holds (M=L-16, K=kc+2..kc+3)
            v2f a = *reinterpret_cast<const v2f*>(&hsh[l15 * KCH + kc]);
            // B: 4x16 f32 mirrored layout, read straight from W (L2).
            v2f bm = *reinterpret_cast<const v2f*>(wr + kc);
            acc = __builtin_amdgcn_wmma_f32_16x16x4_f32(
                false, a, false, bm, (short)0, acc, false, false);
        }
    }

    // ---- store C (bias already folded into init) ----
    const size_t rowbase = (size_t)bidx * SEQ_N + (size_t)n0 + ((size_t)hi << 3);
    float* op = out + rowbase * OUTF + obase + l15;
    #pragma unroll
    for (int r = 0; r < 8; ++r)
        op[(size_t)r * OUTF] = acc[r];
}

extern "C" void kernel_launch(void* const* d_in, const int* in_sizes, int n_in,
                              void* d_out, int out_size, void* d_ws, size_t ws_size,
                              hipStream_t stream) {
    (void)in_sizes; (void)n_in; (void)out_size; (void)d_ws; (void)ws_size;
    const float* x    = (const float*)d_in[0];
    const int*   nbh  = (const int*)  d_in[1];
    const float* dst  = (const float*)d_in[2];
    const float* W    = (const float*)d_in[3];
    const float* bias = (const float*)d_in[4];
    float* out = (float*)d_out;

    dim3 grid(SEQ_N / MT, BATCH);   // (125, 32)
    kcnn_wmma_f32<<<grid, dim3(256), 0, stream>>>(x, nbh, dst, W, bias, out);
}